// GS_39496519254558
// MI455X (gfx1250) — compile-verified
//
#include <hip/hip_runtime.h>

// GraphSAGE 2-layer, N=100000, D=128, E=1.6M — memory-bound (scatter dominates),
// fp32 WMMA 16x16x4 for the 4 small GEMMs (13 GFLOP total, hidden under memory).

#define D 128

typedef __attribute__((ext_vector_type(2))) float v2f;
typedef __attribute__((ext_vector_type(8))) float v8f;

// ---------------- zero scratch (float4 -> global_store_b128) ----------------
__global__ void GS_zero_kernel(float4* __restrict__ p, size_t n4) {
  size_t i = (size_t)blockIdx.x * blockDim.x + threadIdx.x;
  size_t stride = (size_t)gridDim.x * blockDim.x;
  const float4 z = {0.f, 0.f, 0.f, 0.f};
  for (; i < n4; i += stride) p[i] = z;
}

// ---------------- edge scatter: agg[dst] += feat[src]; cnt[dst] += 1 ----------------
// one wave (32 lanes) per edge; lane handles 4 consecutive floats (128/32 = 4).
// unsafeAtomicAdd -> native non-returning global_atomic_add_f32 (no CAS loop).
__global__ __launch_bounds__(256) void GS_scatter_kernel(
    const float* __restrict__ feat, const int* __restrict__ src,
    const int* __restrict__ dst, float* __restrict__ agg,
    float* __restrict__ cnt, int E) {
  long long tid = (long long)blockIdx.x * blockDim.x + threadIdx.x;
  int e = (int)(tid >> 5);
  int lane = (int)(tid & 31);
  if (e >= E) return;
  // edge index is wave-uniform: force scalar loads via readfirstlane
  int es = __builtin_amdgcn_readfirstlane(e);
  int s = src[es];
  int d = dst[es];
  const float4 v = *reinterpret_cast<const float4*>(feat + (size_t)s * D + lane * 4);
  float* a = agg + (size_t)d * D + lane * 4;
  unsafeAtomicAdd(a + 0, v.x);
  unsafeAtomicAdd(a + 1, v.y);
  unsafeAtomicAdd(a + 2, v.z);
  unsafeAtomicAdd(a + 3, v.w);
  if (cnt != nullptr && lane == 0) unsafeAtomicAdd(cnt + d, 1.0f);
}

// ---------------- fused SAGE layer GEMM ----------------
// out[m, :] = (agg[m,:] / max(cnt[m],1)) @ Wl^T + bl + x[m,:] @ Wr^T   (+ ReLU)
// LDS (floats): Wl^T [128x129], Wr^T [128x129], A tile [16x130], X tile [16x130], bias[128]
#define WSTR 129
#define ASTR 130
#define SM_WL 0
#define SM_WR (SM_WL + D * WSTR)
#define SM_A  (SM_WR + D * WSTR)
#define SM_X  (SM_A + 16 * ASTR)
#define SM_B  (SM_X + 16 * ASTR)
#define SM_TOT (SM_B + D)

__global__ __launch_bounds__(256, 2) void GS_sage_gemm_kernel(
    const float* __restrict__ agg, const float* __restrict__ cnt,
    const float* __restrict__ xin, const float* __restrict__ Wl,
    const float* __restrict__ bl, const float* __restrict__ Wr,
    float* __restrict__ out, int relu, int nrows) {
  extern __shared__ float sm[];
  const int t = threadIdx.x;

  // Stage W transposed once per block: sW[k*WSTR + j] = W[j*D + k].
  // Global reads coalesced over k; LDS writes distributed via odd stride.
  for (int idx = t; idx < D * D; idx += 256) {
    int j = idx >> 7;
    int k = idx & 127;
    sm[SM_WL + k * WSTR + j] = Wl[idx];
    sm[SM_WR + k * WSTR + j] = Wr[idx];
  }
  if (t < D) sm[SM_B + t] = bl[t];
  __syncthreads();

  const int wid  = t >> 5;   // wave id -> N-tile (8 waves cover all 128 cols)
  const int lane = t & 31;
  const int half = lane >> 4;
  const int r    = lane & 15;
  const int j0   = wid * 16;

  const int mtiles = nrows >> 4;  // 6250, exact (100000 % 16 == 0)
  for (int mt = blockIdx.x; mt < mtiles; mt += gridDim.x) {
    const int m0 = mt << 4;
    __syncthreads();  // previous tile's compute done before restaging
    for (int idx = t; idx < 16 * D; idx += 256) {
      int row = idx >> 7;
      int col = idx & 127;
      float c = cnt[m0 + row];
      float inv = 1.0f / fmaxf(c, 1.0f);
      sm[SM_A + row * ASTR + col] = agg[(size_t)(m0 + row) * D + col] * inv;
      sm[SM_X + row * ASTR + col] = xin[(size_t)(m0 + row) * D + col];
    }
    __syncthreads();

    v8f acc = {};
#pragma unroll
    for (int k0 = 0; k0 < D; k0 += 4) {
      v2f a, b;
      // Wl-term, A = scaled aggregate tile
      a.x = sm[SM_A + r * ASTR + k0 + 2 * half];
      a.y = sm[SM_A + r * ASTR + k0 + 2 * half + 1];
      b.x = sm[SM_WL + (k0 + half) * WSTR + j0 + r];
      b.y = sm[SM_WL + (k0 + 2 + half) * WSTR + j0 + r];
      acc = __builtin_amdgcn_wmma_f32_16x16x4_f32(false, a, false, b,
                                                  (short)0, acc, false, false);
      // Wr-term, A = self-feature tile
      a.x = sm[SM_X + r * ASTR + k0 + 2 * half];
      a.y = sm[SM_X + r * ASTR + k0 + 2 * half + 1];
      b.x = sm[SM_WR + (k0 + half) * WSTR + j0 + r];
      b.y = sm[SM_WR + (k0 + 2 + half) * WSTR + j0 + r];
      acc = __builtin_amdgcn_wmma_f32_16x16x4_f32(false, a, false, b,
                                                  (short)0, acc, false, false);
    }

    float bias = sm[SM_B + j0 + r];
#pragma unroll
    for (int v = 0; v < 8; ++v) {
      float o = acc[v] + bias;
      if (relu) o = fmaxf(o, 0.0f);
      // C layout: lanes 0-15 -> M = v, lanes 16-31 -> M = v + 8; N = j0 + r
      out[(size_t)(m0 + v + 8 * half) * D + j0 + r] = o;
    }
  }
}

extern "C" void kernel_launch(void* const* d_in, const int* in_sizes, int n_in,
                              void* d_out, int out_size, void* d_ws, size_t ws_size,
                              hipStream_t stream) {
  const float* x   = (const float*)d_in[0];
  const int*   ei  = (const int*)d_in[1];  // per harness contract: integer -> const int*
  const float* Wl1 = (const float*)d_in[2];
  const float* bl1 = (const float*)d_in[3];
  const float* Wr1 = (const float*)d_in[4];
  const float* Wl2 = (const float*)d_in[5];
  const float* bl2 = (const float*)d_in[6];
  const float* Wr2 = (const float*)d_in[7];

  const int E = in_sizes[1] / 2;     // 1,600,000
  const int n = in_sizes[0] / D;     // 100,000

  const int* src = ei;
  const int* dst = ei + E;

  // workspace: agg [n*D] | cnt [n] | h [n*D]  (~103 MB)
  // n*D is a multiple of 4; cnt region is not padded; (n*D + n)/4 zeroing assumes n % 4 == 0.
  float* agg = (float*)d_ws;
  float* cnt = agg + (size_t)n * D;
  float* h   = cnt + n;
  float* out = (float*)d_out;

  const long long sthreads = (long long)E * 32;
  const int sblocks = (int)((sthreads + 255) / 256);
  const size_t smem = (size_t)SM_TOT * sizeof(float);

  // ---- layer 1 ----
  GS_zero_kernel<<<2048, 256, 0, stream>>>((float4*)agg, ((size_t)n * D + n) / 4);
  GS_scatter_kernel<<<sblocks, 256, 0, stream>>>(x, src, dst, agg, cnt, E);
  GS_sage_gemm_kernel<<<512, 256, smem, stream>>>(agg, cnt, x, Wl1, bl1, Wr1,
                                                  h, /*relu=*/1, n);
  // ---- layer 2 (degree counts unchanged; re-zero agg only) ----
  GS_zero_kernel<<<2048, 256, 0, stream>>>((float4*)agg, (size_t)n * D / 4);
  GS_scatter_kernel<<<sblocks, 256, 0, stream>>>(h, src, dst, agg, nullptr, E);
  GS_sage_gemm_kernel<<<512, 256, smem, stream>>>(agg, cnt, h, Wl2, bl2, Wr2,
                                                  out, /*relu=*/0, n);
}